// OhemCrossEntropy_69767448756300
// MI455X (gfx1250) — compile-verified
//
#include <hip/hip_runtime.h>
#include <hip/hip_bf16.h>
#include <cstdint>

#define IGNORE_LABEL 255
#define THRESH_F     0.7f
#define MIN_KEPT     100000
#define BATCH        8
#define NCLS         19
#define HW           (512 * 512)
#define NPIX         (BATCH * HW)        // 2,097,152
#define HBINS        65536

typedef __attribute__((ext_vector_type(4))) float f4;

__device__ __forceinline__ f4 vmax4(f4 a, f4 b) {
    f4 r; r.x = fmaxf(a.x, b.x); r.y = fmaxf(a.y, b.y);
    r.z = fmaxf(a.z, b.z); r.w = fmaxf(a.w, b.w); return r;
}

// ---------------- K0: zero histograms + control block ----------------
__global__ void k_init(unsigned int* __restrict__ p, int n) {
    int i = blockIdx.x * blockDim.x + threadIdx.x;
    if (i < n) p[i] = 0u;
}

// ---------------- K1: streaming softmax pass ----------------
// 256 threads/block, 4 pixels/thread -> 1024 pixels/block, 2048 blocks.
// Emits logp[NPIX], hi16-bit histogram of pred over valid pixels, n_valid.
__global__ __launch_bounds__(256)
void k_main(const float* __restrict__ score, const int* __restrict__ target,
            float* __restrict__ logp, unsigned int* __restrict__ hist1,
            unsigned int* __restrict__ ctrl) {
    __shared__ int tgt_lds[1024];
    __shared__ unsigned int blk_cnt;
    const int tid = threadIdx.x;
    if (tid == 0) blk_cnt = 0u;

    const int px0 = blockIdx.x * 1024 + tid * 4;

    // Async copy this block's 1024 labels (4 per lane, 16B each) into LDS.
    {
        unsigned long long gaddr =
            (unsigned long long)(const void*)(target + blockIdx.x * 1024 + tid * 4);
        unsigned int lofs = (unsigned int)(uintptr_t)(&tgt_lds[tid * 4]);
        asm volatile("global_load_async_to_lds_b128 %0, %1, off"
                     :: "v"(lofs), "v"(gaddr) : "memory");
    }

    // Stream 19 channels for 4 consecutive pixels (NT: single-use data).
    const int b  = px0 / HW;
    const int hw = px0 % HW;
    const float* base = score + (size_t)b * NCLS * HW + hw;
    f4 v[NCLS];
#pragma unroll
    for (int c = 0; c < NCLS; ++c)
        v[c] = __builtin_nontemporal_load((const f4*)(base + (size_t)c * HW));

    f4 m = v[0];
#pragma unroll
    for (int c = 1; c < NCLS; ++c) m = vmax4(m, v[c]);

    f4 s = {0.f, 0.f, 0.f, 0.f};
#pragma unroll
    for (int c = 0; c < NCLS; ++c) {
        s.x += expf(v[c].x - m.x); s.y += expf(v[c].y - m.y);
        s.z += expf(v[c].z - m.z); s.w += expf(v[c].w - m.w);
    }

    // Wait for async label copy, then gather target-class logits.
    asm volatile("s_wait_asynccnt 0x0" ::: "memory");
    __syncthreads();
    int t0 = tgt_lds[tid * 4 + 0], t1 = tgt_lds[tid * 4 + 1];
    int t2 = tgt_lds[tid * 4 + 2], t3 = tgt_lds[tid * 4 + 3];
    int c0 = (t0 == IGNORE_LABEL) ? 0 : t0;
    int c1 = (t1 == IGNORE_LABEL) ? 0 : t1;
    int c2 = (t2 == IGNORE_LABEL) ? 0 : t2;
    int c3 = (t3 == IGNORE_LABEL) ? 0 : t3;

    f4 st = {0.f, 0.f, 0.f, 0.f};
#pragma unroll
    for (int c = 0; c < NCLS; ++c) {
        st.x = (c == c0) ? v[c].x : st.x;
        st.y = (c == c1) ? v[c].y : st.y;
        st.z = (c == c2) ? v[c].z : st.z;
        st.w = (c == c3) ? v[c].w : st.w;
    }

    f4 lp;
    lp.x = st.x - m.x - logf(s.x);
    lp.y = st.y - m.y - logf(s.y);
    lp.z = st.z - m.z - logf(s.z);
    lp.w = st.w - m.w - logf(s.w);
    *(f4*)(logp + px0) = lp;

    // Radix-histogram (hi 16 bits of positive-float key) + valid count.
    unsigned int lc = 0;
    if (t0 != IGNORE_LABEL) { atomicAdd(&hist1[__float_as_uint(expf(lp.x)) >> 16], 1u); ++lc; }
    if (t1 != IGNORE_LABEL) { atomicAdd(&hist1[__float_as_uint(expf(lp.y)) >> 16], 1u); ++lc; }
    if (t2 != IGNORE_LABEL) { atomicAdd(&hist1[__float_as_uint(expf(lp.z)) >> 16], 1u); ++lc; }
    if (t3 != IGNORE_LABEL) { atomicAdd(&hist1[__float_as_uint(expf(lp.w)) >> 16], 1u); ++lc; }
    atomicAdd(&blk_cnt, lc);
    __syncthreads();
    if (tid == 0) atomicAdd(&ctrl[0], blk_cnt);
}

// ---------------- K2/K4: single-block rank selection over 65536 bins ----------------
// phase 0: rank = min(MIN_KEPT, n_valid-1); writes ctrl[1]=hi-bin, ctrl[2]=rank-in-bin
// phase 1: rank = ctrl[2]; writes ctrl[3] = bits(max(kth_value, 0.7))
__global__ __launch_bounds__(1024)
void k_select(const unsigned int* __restrict__ hist, unsigned int* __restrict__ ctrl,
              int phase) {
    __shared__ unsigned int part[1024];
    const int tid  = threadIdx.x;
    const int base = tid * 64;
    unsigned int local = 0;
    for (int i = 0; i < 64; ++i) local += hist[base + i];
    part[tid] = local;
    __syncthreads();
    for (int off = 1; off < 1024; off <<= 1) {
        unsigned int add = (tid >= off) ? part[tid - off] : 0u;
        __syncthreads();
        part[tid] += add;
        __syncthreads();
    }
    unsigned int incl = part[tid];
    unsigned int excl = incl - local;

    unsigned int rank;
    if (phase == 0) {
        unsigned int nv = ctrl[0];
        if (nv == 0u) {                       // no valid pixels: threshold = +inf
            if (tid == 0) { ctrl[1] = 0xFFFFFFFFu; ctrl[3] = 0x7F800000u; }
            return;
        }
        unsigned int k = nv - 1u;
        rank = k < (unsigned int)MIN_KEPT ? k : (unsigned int)MIN_KEPT;
    } else {
        if (ctrl[1] == 0xFFFFFFFFu) return;
        rank = ctrl[2];
    }

    if (excl <= rank && rank < incl) {
        unsigned int c = excl;
        for (int i = 0; i < 64; ++i) {
            unsigned int h = hist[base + i];
            if (rank < c + h) {
                if (phase == 0) { ctrl[1] = (unsigned int)(base + i); ctrl[2] = rank - c; }
                else {
                    unsigned int bits = (ctrl[1] << 16) | (unsigned int)(base + i);
                    float mv = __uint_as_float(bits);
                    ctrl[3] = __float_as_uint(fmaxf(mv, THRESH_F));
                }
                break;
            }
            c += h;
        }
    }
}

// ---------------- K3: low-16-bit refinement histogram ----------------
__global__ __launch_bounds__(256)
void k_refine(const float* __restrict__ logp, const int* __restrict__ target,
              const unsigned int* __restrict__ ctrl, unsigned int* __restrict__ hist2) {
    unsigned int bstar = ctrl[1];
    if (bstar == 0xFFFFFFFFu) return;
    int i0 = (blockIdx.x * 256 + threadIdx.x) * 4;
#pragma unroll
    for (int j = 0; j < 4; ++j) {
        int t = target[i0 + j];
        if (t != IGNORE_LABEL) {
            unsigned int u = __float_as_uint(expf(logp[i0 + j]));
            if ((u >> 16) == bstar) atomicAdd(&hist2[u & 0xFFFFu], 1u);
        }
    }
}

// ---------------- K5: masked sum / count reduction ----------------
__global__ __launch_bounds__(256)
void k_final(const float* __restrict__ logp, const int* __restrict__ target,
             unsigned int* __restrict__ ctrl) {
    __shared__ float        sred[256];
    __shared__ unsigned int cred[256];
    const int tid = threadIdx.x;
    const float thr = __uint_as_float(ctrl[3]);
    int i0 = (blockIdx.x * 256 + tid) * 4;
    float s = 0.f; unsigned int c = 0;
#pragma unroll
    for (int j = 0; j < 4; ++j) {
        int t = target[i0 + j];
        if (t != IGNORE_LABEL) {
            float lp = logp[i0 + j];
            if (expf(lp) < thr) { s += -lp; ++c; }
        }
    }
    sred[tid] = s; cred[tid] = c;
    __syncthreads();
    for (int off = 128; off > 0; off >>= 1) {
        if (tid < off) { sred[tid] += sred[tid + off]; cred[tid] += cred[tid + off]; }
        __syncthreads();
    }
    if (tid == 0) {
        atomicAdd((float*)&ctrl[4], sred[0]);
        atomicAdd(&ctrl[5], cred[0]);
    }
}

__global__ void k_write(const unsigned int* __restrict__ ctrl, float* __restrict__ out) {
    if (threadIdx.x == 0 && blockIdx.x == 0) {
        float s = __uint_as_float(ctrl[4]);
        unsigned int c = ctrl[5];
        out[0] = 1.0f * s / (float)(c > 0u ? c : 1u);   // LOSS_WEIGHT = 1.0
    }
}

// ---------------- host launcher ----------------
extern "C" void kernel_launch(void* const* d_in, const int* in_sizes, int n_in,
                              void* d_out, int out_size, void* d_ws, size_t ws_size,
                              hipStream_t stream) {
    const float* score  = (const float*)d_in[0];
    const int*   target = (const int*)d_in[1];
    float*       out    = (float*)d_out;

    // ws layout (u32 units): logp[NPIX] | hist1[65536] | hist2[65536] | ctrl[8]
    unsigned int* ws    = (unsigned int*)d_ws;
    float*        logp  = (float*)ws;
    unsigned int* hist1 = ws + NPIX;
    unsigned int* hist2 = ws + NPIX + HBINS;
    unsigned int* ctrl  = ws + NPIX + 2 * HBINS;

    const int nzero = 2 * HBINS + 8;
    k_init  <<<(nzero + 255) / 256, 256, 0, stream>>>(hist1, nzero);
    k_main  <<<NPIX / 1024, 256, 0, stream>>>(score, target, logp, hist1, ctrl);
    k_select<<<1, 1024, 0, stream>>>(hist1, ctrl, 0);
    k_refine<<<NPIX / 1024, 256, 0, stream>>>(logp, target, ctrl, hist2);
    k_select<<<1, 1024, 0, stream>>>(hist2, ctrl, 1);
    k_final <<<NPIX / 1024, 256, 0, stream>>>(logp, target, ctrl);
    k_write <<<1, 1, 0, stream>>>(ctrl, out);
}